// DNN_Beamformer_77627238908491
// MI455X (gfx1250) — compile-verified
//
#include <hip/hip_runtime.h>
#include <math.h>

#define Bb 8
#define Ff 257
#define Cc 6
#define Tt 1500
#define ATTD 320
#define MROWS (Bb*Cc)         /* 48  */
#define KDIM  Ff              /* 257 */
#define NTILES (ATTD/16)      /* 20  */
#define MTILES (MROWS/16)     /* 3   */
#define KSTEPS 65             /* K padded to 260, 4 per WMMA */
#define BFT (Bb*Ff*Tt)        /* 3084000 */

#define APACK_N (MTILES*KSTEPS*32)   /* 6240  v2f */
#define BPACK_N (NTILES*KSTEPS*32)   /* 41600 v2f */

typedef __attribute__((ext_vector_type(2))) float v2f;
typedef __attribute__((ext_vector_type(4))) float v4f;
typedef __attribute__((ext_vector_type(8))) float v8f;

// ---------------- Kernel A: psd_feat (B*C, F) = |offdiag row sum| / (C-1) ----
__global__ void feat_kernel(const float* __restrict__ psd_s_r,
                            const float* __restrict__ psd_s_i,
                            float* __restrict__ feat) {
    int id = blockIdx.x * blockDim.x + threadIdx.x;
    if (id >= MROWS * Ff) return;
    int f = id % Ff;
    int c = (id / Ff) % Cc;
    int b = id / (Ff * Cc);
    const int base = ((b * Ff + f) * Cc + c) * Cc;
    float sr = 0.f, si = 0.f;
#pragma unroll
    for (int e = 0; e < Cc; ++e) {
        if (e != c) { sr += psd_s_r[base + e]; si += psd_s_i[base + e]; }
    }
    feat[id] = sqrtf(sr * sr + si * si) * (1.0f / (float)(Cc - 1));
}

// ------- Kernel A2: pack WMMA A/B fragments (zero-padded, per-lane layout) ---
// A fragment (16x4 f32): lane L: M = m0 + (L&15); vgpr j holds K = 4*kk + (L>>4)*2 + j
// B fragment (4x16 f32): lane L: N = n0 + (L&15); vgpr j holds K = 4*kk + (L>>4)*2 + j
__global__ void pack_frags_kernel(const float* __restrict__ feat,
                                  const float* __restrict__ w_mlp,
                                  v2f* __restrict__ Apack,
                                  v2f* __restrict__ Bpack) {
    int id = blockIdx.x * blockDim.x + threadIdx.x;
    if (id < APACK_N) {
        int lane = id & 31;
        int kk   = (id >> 5) % KSTEPS;
        int mt   = id / (32 * KSTEPS);
        int row = lane & 15, hi = lane >> 4;
        int k = kk * 4 + hi * 2;
        int m = mt * 16 + row;
        v2f a;
        a.x = (k     < KDIM) ? feat[m * KDIM + k]     : 0.f;
        a.y = (k + 1 < KDIM) ? feat[m * KDIM + k + 1] : 0.f;
        Apack[id] = a;
        return;
    }
    int id2 = id - APACK_N;
    if (id2 < BPACK_N) {
        int lane = id2 & 31;
        int kk   = (id2 >> 5) % KSTEPS;
        int nt   = id2 / (32 * KSTEPS);
        int row = lane & 15, hi = lane >> 4;
        int k = kk * 4 + hi * 2;
        int n = nt * 16 + row;
        v2f bfr;
        bfr.x = (k     < KDIM) ? w_mlp[(k)     * ATTD + n] : 0.f;
        bfr.y = (k + 1 < KDIM) ? w_mlp[(k + 1) * ATTD + n] : 0.f;
        Bpack[id2] = bfr;
    }
}

// ------------- Kernel B: WMMA f32 GEMM, branch-free inner loop ---------------
__global__ void mlp_wmma_kernel(const v2f* __restrict__ Apack,
                                const v2f* __restrict__ Bpack,
                                const float* __restrict__ b_mlp,
                                const float* __restrict__ w_gvec,
                                float* __restrict__ epart) {
    const int tile = blockIdx.x;
    const int mt = tile / NTILES, nt = tile % NTILES;
    const int lane = threadIdx.x;      // 0..31, EXEC all ones throughout loop
    const int row  = lane & 15;
    const int hi   = lane >> 4;

    const v2f* __restrict__ Ap = Apack + mt * (KSTEPS * 32) + lane;
    const v2f* __restrict__ Bp = Bpack + nt * (KSTEPS * 32) + lane;

    v8f acc = {};
#pragma unroll 5
    for (int kk = 0; kk < KSTEPS; ++kk) {
        v2f a   = Ap[kk * 32];   // coalesced global_load_b64
        v2f bfr = Bp[kk * 32];   // coalesced global_load_b64
        acc = __builtin_amdgcn_wmma_f32_16x16x4_f32(false, a, false, bfr,
                                                    (short)0, acc, false, false);
    }

    const int n0 = nt * 16, m0 = mt * 16;
    const float bias = b_mlp[n0 + row];
    const float wg   = w_gvec[n0 + row];
#pragma unroll
    for (int v = 0; v < 8; ++v) {
        // D layout: vgpr v, lanes 0-15 -> M=m0+v, lanes 16-31 -> M=m0+v+8; N=n0+row
        float val = tanhf(acc[v] + bias) * wg;
#pragma unroll
        for (int off = 8; off >= 1; off >>= 1)
            val += __shfl_xor(val, off, 32);   // reduce within each 16-lane half
        int m = m0 + v + hi * 8;
        if (row == 0) epart[m * NTILES + nt] = val;
    }
}

// ---------------- Kernel C: reduce partials, add b_gvec, softmax(2e) ---------
__global__ void softmax_kernel(const float* __restrict__ epart,
                               const float* __restrict__ b_gvec,
                               float* __restrict__ ubuf,
                               float* __restrict__ out_u) {
    __shared__ float es[MROWS];
    int t = threadIdx.x;
    if (t < MROWS) {
        float s = 0.f;
        for (int j = 0; j < NTILES; ++j) s += epart[t * NTILES + j];
        es[t] = 2.0f * (s + b_gvec[0]);        // scaling * (e + b_gvec)
    }
    __syncthreads();
    if (t < Bb) {
        float mx = -1e30f;
#pragma unroll
        for (int c = 0; c < Cc; ++c) mx = fmaxf(mx, es[t * Cc + c]);
        float ex[Cc]; float den = 0.f;
#pragma unroll
        for (int c = 0; c < Cc; ++c) { ex[c] = __expf(es[t * Cc + c] - mx); den += ex[c]; }
#pragma unroll
        for (int c = 0; c < Cc; ++c) {
            float u = ex[c] / den;
            ubuf[t * Cc + c]  = u;
            out_u[t * Cc + c] = u;
        }
    }
}

// -------- Kernel D: per (b,f) complex 6x6 solve, trace-normalize, steer ------
__global__ void mvdr_solve_kernel(const float* __restrict__ psd_s_r,
                                  const float* __restrict__ psd_s_i,
                                  const float* __restrict__ psd_n_r,
                                  const float* __restrict__ psd_n_i,
                                  const float* __restrict__ ubuf,
                                  float* __restrict__ bfc) {
    int bf = blockIdx.x * blockDim.x + threadIdx.x;
    if (bf >= Bb * Ff) return;
    const int b = bf / Ff;
    float Ar[Cc][Cc], Ai[Cc][Cc], Xr[Cc][Cc], Xi[Cc][Cc];
    const int base = bf * Cc * Cc;
#pragma unroll
    for (int i = 0; i < Cc; ++i)
#pragma unroll
        for (int j = 0; j < Cc; ++j) {
            Ar[i][j] = psd_n_r[base + i * Cc + j];
            Ai[i][j] = psd_n_i[base + i * Cc + j];
            Xr[i][j] = psd_s_r[base + i * Cc + j];
            Xi[i][j] = psd_s_i[base + i * Cc + j];
        }
    // Gauss-Jordan: X <- A^{-1} X  (A = psd_n is PSD + I: no pivoting needed)
#pragma unroll
    for (int k = 0; k < Cc; ++k) {
        float pr = Ar[k][k], pi = Ai[k][k];
        float inv = 1.0f / (pr * pr + pi * pi);
        float ir = pr * inv, ii = -pi * inv;
#pragma unroll
        for (int j = 0; j < Cc; ++j) {
            float tr = Ar[k][j], ti = Ai[k][j];
            Ar[k][j] = tr * ir - ti * ii;  Ai[k][j] = tr * ii + ti * ir;
            tr = Xr[k][j]; ti = Xi[k][j];
            Xr[k][j] = tr * ir - ti * ii;  Xi[k][j] = tr * ii + ti * ir;
        }
#pragma unroll
        for (int i = 0; i < Cc; ++i) {
            if (i == k) continue;
            float fr = Ar[i][k], fi = Ai[i][k];
#pragma unroll
            for (int j = 0; j < Cc; ++j) {
                float akr = Ar[k][j], aki = Ai[k][j];
                Ar[i][j] -= fr * akr - fi * aki;
                Ai[i][j] -= fr * aki + fi * akr;
                float xkr = Xr[k][j], xki = Xi[k][j];
                Xr[i][j] -= fr * xkr - fi * xki;
                Xi[i][j] -= fr * xki + fi * xkr;
            }
        }
    }
    float trr = 1e-6f, tri = 0.f;   // trace + eps (eps on real part)
#pragma unroll
    for (int c = 0; c < Cc; ++c) { trr += Xr[c][c]; tri += Xi[c][c]; }
    float invd = 1.0f / (trr * trr + tri * tri);
    float sr = trr * invd, si = -tri * invd;   // 1/(trace+eps)
#pragma unroll
    for (int e = 0; e < Cc; ++e) {
        float br = 0.f, bi = 0.f;
#pragma unroll
        for (int c = 0; c < Cc; ++c) {
            float uc = ubuf[b * Cc + c];
            br += Xr[e][c] * uc;
            bi += Xi[e][c] * uc;
        }
        float vr = br * sr - bi * si;
        float vi = br * si + bi * sr;
        bfc[(bf * Cc + e) * 2 + 0] =  vr;   // conj(bf_vec)
        bfc[(bf * Cc + e) * 2 + 1] = -vi;
    }
}

// -------- Kernel E: bandwidth-bound apply: enh = h^H x, NT float4 streaming --
__global__ void apply_kernel(const float* __restrict__ data_r,
                             const float* __restrict__ data_i,
                             const float* __restrict__ bfc,
                             float* __restrict__ out_r,
                             float* __restrict__ out_i) {
    const int bf = blockIdx.x;
    float hr[Cc], hi[Cc];
#pragma unroll
    for (int c = 0; c < Cc; ++c) {
        hr[c] = bfc[(bf * Cc + c) * 2 + 0];
        hi[c] = bfc[(bf * Cc + c) * 2 + 1];
    }
    const long baseC = (long)bf * Cc * Tt;
    const long baseT = (long)bf * Tt;
    for (int t4 = threadIdx.x; t4 < Tt / 4; t4 += blockDim.x) {
        v4f ar = {}, ai = {};
#pragma unroll
        for (int c = 0; c < Cc; ++c) {
            // streamed exactly once: non-temporal loads (TH=NT) keep L2 clean
            v4f xr = __builtin_nontemporal_load((const v4f*)(data_r + baseC + (long)c * Tt + t4 * 4));
            v4f xi = __builtin_nontemporal_load((const v4f*)(data_i + baseC + (long)c * Tt + t4 * 4));
            ar += hr[c] * xr - hi[c] * xi;   // Re(h*x)
            ai += hr[c] * xi + hi[c] * xr;   // Im(h*x)
        }
        __builtin_nontemporal_store(ar, (v4f*)(out_r + baseT + t4 * 4));
        __builtin_nontemporal_store(ai, (v4f*)(out_i + baseT + t4 * 4));
    }
}

extern "C" void kernel_launch(void* const* d_in, const int* in_sizes, int n_in,
                              void* d_out, int out_size, void* d_ws, size_t ws_size,
                              hipStream_t stream) {
    const float* data_r  = (const float*)d_in[0];
    const float* data_i  = (const float*)d_in[1];
    const float* psd_s_r = (const float*)d_in[2];
    const float* psd_s_i = (const float*)d_in[3];
    const float* psd_n_r = (const float*)d_in[4];
    const float* psd_n_i = (const float*)d_in[5];
    const float* w_mlp   = (const float*)d_in[6];
    const float* b_mlp   = (const float*)d_in[7];
    const float* w_gvec  = (const float*)d_in[8];
    const float* b_gvec  = (const float*)d_in[9];
    (void)in_sizes; (void)n_in; (void)ws_size;

    float* ws    = (float*)d_ws;
    float* feat  = ws;                            // 12336 floats
    v2f*   Apack = (v2f*)(feat + MROWS * Ff);     // 6240  v2f (12480 floats)
    v2f*   Bpack = Apack + APACK_N;               // 41600 v2f (83200 floats)
    float* epart = (float*)(Bpack + BPACK_N);     // 960 floats
    float* ubuf  = epart + MROWS * NTILES;        // 48
    float* bfc   = ubuf + MROWS;                  // 24672

    float* out_r = (float*)d_out;                 // (B,F,T)
    float* out_i = out_r + BFT;                   // (B,F,T)
    float* out_u = out_r + 2 * BFT;               // (B,C)

    // A: features
    {
        int n = MROWS * Ff;
        feat_kernel<<<(n + 255) / 256, 256, 0, stream>>>(psd_s_r, psd_s_i, feat);
    }
    // A2: pack WMMA fragments (zero-padded)
    {
        int n = APACK_N + BPACK_N;
        pack_frags_kernel<<<(n + 255) / 256, 256, 0, stream>>>(feat, w_mlp, Apack, Bpack);
    }
    // B: branch-free WMMA f32 GEMM (one wave per 16x16 tile)
    mlp_wmma_kernel<<<MTILES * NTILES, 32, 0, stream>>>(Apack, Bpack, b_mlp, w_gvec, epart);
    // C: reduce + softmax -> u
    softmax_kernel<<<1, 64, 0, stream>>>(epart, b_gvec, ubuf, out_u);
    // D: 2056 complex 6x6 solves -> conj(bf_vec)
    {
        int n = Bb * Ff;
        mvdr_solve_kernel<<<(n + 63) / 64, 64, 0, stream>>>(psd_s_r, psd_s_i,
                                                            psd_n_r, psd_n_i,
                                                            ubuf, bfc);
    }
    // E: bandwidth-bound apply
    apply_kernel<<<Bb * Ff, 256, 0, stream>>>(data_r, data_i, bfc, out_r, out_i);
}